// PellLucasTimeSpine_77300821393851
// MI455X (gfx1250) — compile-verified
//
#include <hip/hip_runtime.h>
#include <hip/hip_bf16.h>

typedef float v2f __attribute__((ext_vector_type(2)));
typedef float v4f __attribute__((ext_vector_type(4)));
typedef float v8f __attribute__((ext_vector_type(8)));

#define DM 1024          // D_MODEL (reference constant)
#define MAXD 20          // MAX_DEPTH
#define NSPMAX 16

// ---------------------------------------------------------------------------
// Kernel 1: build the 16 x 1024 table with one workgroup.
//   Phase A (16 lanes): walk each chain, build visit mask + count,
//                       write A = rsqrt(count) * visit_mask into LDS (16x16 f32).
//   Phase B (8 waves):  table = A x emb via V_WMMA_F32_16X16X4_F32,
//                       64 column tiles, 4 K-steps each.
// ---------------------------------------------------------------------------
__global__ void __launch_bounds__(256)
spine_build_table(const float* __restrict__ emb,
                  const int* __restrict__ spine,
                  int nsp,
                  float* __restrict__ table) {
    __shared__ float Msh[16][16];
    __shared__ int   s_spine[NSPMAX];
    __shared__ int   s_parent[NSPMAX];

    const int tid = threadIdx.x;

    // zero A matrix (covers rows/cols >= nsp too)
    ((float*)Msh)[tid] = 0.0f;
    if (tid < NSPMAX) {
        int s = (tid < nsp) ? spine[tid] : 0;
        s_spine[tid]  = s;
        s_parent[tid] = (int)(((long long)s * 1000LL) / 2414LL);
    }
    __syncthreads();

    if (tid < nsp && tid < NSPMAX) {
        int idx = tid;
        int cnt = 0;
        unsigned mask = 0u;
        for (int step = 0; step < MAXD; ++step) {
            mask |= (1u << idx);
            ++cnt;
            if (s_spine[idx] <= 0) break;               // path terminates at spine==0
            int cur = s_parent[idx];
            int k = 0;                                   // searchsorted(right): #entries <= cur
            for (int j = 0; j < nsp; ++j) k += (s_spine[j] <= cur) ? 1 : 0;
            idx = k - 1;
        }
        float scale = rsqrtf((float)cnt);                // fold 1/sqrt(len) into A
        for (int j = 0; j < NSPMAX; ++j)
            Msh[tid][j] = ((mask >> j) & 1u) ? scale : 0.0f;
    }
    __syncthreads();

    // ---- WMMA phase: D(16x16 tile) = A(16x16) x B(16x16 tile of emb) ----
    const int lane = tid & 31;
    const int wv   = tid >> 5;          // 8 waves
    const int row  = lane & 15;
    const int hi   = lane >> 4;         // 0: K lo-half, 1: K hi-half

    // A fragments for the 4 K-steps (K = 4*k0 .. 4*k0+3)
    v2f a[4];
#pragma unroll
    for (int k0 = 0; k0 < 4; ++k0) {
        int kb = k0 * 4 + hi * 2;
        a[k0].x = Msh[row][kb];
        a[k0].y = Msh[row][kb + 1];
    }

    const int nspc = (nsp < NSPMAX) ? nsp : NSPMAX;
    for (int t = wv; t < DM / 16; t += 8) {
        const int col = t * 16 + row;
        v8f c = {0.f, 0.f, 0.f, 0.f, 0.f, 0.f, 0.f, 0.f};
#pragma unroll
        for (int k0 = 0; k0 < 4; ++k0) {
            int kr  = k0 * 4 + hi * 2;
            int kr0 = (kr     < nspc) ? kr     : nspc - 1;  // clamp (A is 0 there anyway)
            int kr1 = (kr + 1 < nspc) ? kr + 1 : nspc - 1;
            v2f b;
            b.x = emb[kr0 * DM + col];
            b.y = emb[kr1 * DM + col];
            c = __builtin_amdgcn_wmma_f32_16x16x4_f32(
                    /*neg_a=*/false, a[k0], /*neg_b=*/false, b,
                    /*c_mod=*/(short)0, c, /*reuse_a=*/false, /*reuse_b=*/false);
        }
#pragma unroll
        for (int v = 0; v < 8; ++v) {
            int r = v + hi * 8;                  // D layout: lanes16-31 hold M=8..15
            table[r * DM + col] = c[v];
        }
    }
}

// ---------------------------------------------------------------------------
// Kernel 2: gather-broadcast. One wave per position; 8 x b128 non-temporal
// stores per lane stream the 4 KB row. Table reads stay cached (64 KB total).
// ---------------------------------------------------------------------------
__global__ void __launch_bounds__(256)
spine_gather(const int* __restrict__ positions,
             const int* __restrict__ spine,
             int nsp,
             const float* __restrict__ table,
             float* __restrict__ out,
             int npos) {
    __shared__ int s_spine[NSPMAX];
    const int tid = threadIdx.x;
    if (tid < NSPMAX)
        s_spine[tid] = (tid < nsp) ? spine[tid] : 0x7fffffff;  // pad -> never counted
    __syncthreads();

    const int wv   = tid >> 5;
    const int lane = tid & 31;
    const int p    = blockIdx.x * 8 + wv;
    if (p >= npos) return;

    const int pos = positions[p];      // uniform per wave -> scalar load
    int k = 0;
#pragma unroll
    for (int j = 0; j < NSPMAX; ++j) k += (s_spine[j] <= pos) ? 1 : 0;
    const int idx = k - 1;             // searchsorted(right) - 1

    const v4f* __restrict__ src = (const v4f*)(table + (size_t)idx * DM);
    v4f* __restrict__       dst = (v4f*)(out + (size_t)p * DM);
#pragma unroll
    for (int i = 0; i < DM / 4 / 32; ++i) {      // 8 iterations of b128
        v4f v = src[lane + i * 32];
        __builtin_nontemporal_store(v, &dst[lane + i * 32]);
    }
}

extern "C" void kernel_launch(void* const* d_in, const int* in_sizes, int n_in,
                              void* d_out, int out_size, void* d_ws, size_t ws_size,
                              hipStream_t stream) {
    const float* emb       = (const float*)d_in[0];   // [nsp, 1024] f32
    const int*   positions = (const int*)d_in[1];     // [4*8192] i32
    const int*   spine     = (const int*)d_in[2];     // [nsp] i32
    const int    npos = in_sizes[1];
    const int    nsp  = in_sizes[2];

    float* table = (float*)d_ws;                      // 16*1024 f32 = 64 KB scratch
    float* out   = (float*)d_out;

    spine_build_table<<<1, 256, 0, stream>>>(emb, spine, nsp, table);

    const int blocks = (npos + 7) / 8;                // 8 positions per block (1 wave each)
    spine_gather<<<blocks, 256, 0, stream>>>(positions, spine, nsp, table, out, npos);
}